// AdditiveAttention_57234734186572
// MI455X (gfx1250) — compile-verified
//
#include <hip/hip_runtime.h>
#include <hip/hip_bf16.h>
#include <stdint.h>

// Problem constants (from reference)
#define B_     8
#define LQ_    256
#define LK_    256
#define QSIZE_ 512
#define KSIZE_ 512
#define HID_   256

typedef float v2f __attribute__((ext_vector_type(2)));
typedef float v4f __attribute__((ext_vector_type(4)));
typedef float v8f __attribute__((ext_vector_type(8)));

// Native gfx1250 tanh. Prefer the builtin (compiler knows the TRANS hazard
// rules); otherwise emit v_tanh_f32 directly. In the asm path, append v_nop:
// ISA §7.4(3) — a TRANS result needs 1 independent op before a dependent
// co-executing VALU may consume it, and the compiler can't see inside asm.
__device__ __forceinline__ float fast_tanhf(float x) {
#if __has_builtin(__builtin_amdgcn_tanhf)
    return __builtin_amdgcn_tanhf(x);
#elif __has_builtin(__builtin_amdgcn_tanh_f32)
    return __builtin_amdgcn_tanh_f32(x);
#else
    float r;
    asm volatile("v_tanh_f32 %0, %1\n\tv_nop" : "=v"(r) : "v"(x));
    return r;
#endif
}

// ---------------------------------------------------------------------------
// Generic f32 GEMM tile kernel: C[M x N] = A[M x K] * W[K x N]
// One 16x16 output tile per wave via V_WMMA_F32_16X16X4_F32 (full f32).
// TRANSPOSED=false: out[row*N + col]
// TRANSPOSED=true : out is KT[b][h][k] (b=row/256, k=row%256, h=col) so the
//                   downstream energy kernel gets coalesced K reads.
// ---------------------------------------------------------------------------
template <bool TRANSPOSED>
__global__ void proj_gemm(const float* __restrict__ A, const float* __restrict__ W,
                          float* __restrict__ out, int K, int N) {
    const int lane = threadIdx.x & 31;
    const int wave = threadIdx.x >> 5;
    const int lo   = lane & 15;    // M (A) / N (B,C)
    const int hi   = lane >> 4;    // K-half select
    const int row0 = blockIdx.x * 16;
    const int col0 = (blockIdx.y * 4 + wave) * 16;

    // A lane view: row = row0+lo, K = k0 + 2*hi + {0,1}  -> contiguous float2
    const float* Arow = A + (size_t)(row0 + lo) * K + 2 * hi;
    // B lane view: col = col0+lo, K = k0 + 2*hi + {0,1}
    const float* Wp = W + (size_t)(2 * hi) * N + col0 + lo;

    v8f acc = {};
#pragma unroll 8
    for (int k0 = 0; k0 < K; k0 += 4) {
        v2f a = *(const v2f*)(Arow + k0);
        v2f b;
        b.x = Wp[(size_t)k0 * N];
        b.y = Wp[(size_t)(k0 + 1) * N];
        acc = __builtin_amdgcn_wmma_f32_16x16x4_f32(
            /*neg_a=*/false, a, /*neg_b=*/false, b,
            /*c_mod=*/(short)0, acc, /*reuse_a=*/false, /*reuse_b=*/false);
    }

    if (!TRANSPOSED) {
        float* o = out + (size_t)(row0 + 8 * hi) * N + col0 + lo;
#pragma unroll
        for (int r = 0; r < 8; ++r) o[(size_t)r * N] = acc[r];
    } else {
        const int rbase = row0 + 8 * hi;
        const int h = col0 + lo;
#pragma unroll
        for (int r = 0; r < 8; ++r) {
            const int rg = rbase + r;
            const int b = rg >> 8;       // LK_ == 256 rows per batch
            const int k = rg & 255;
            out[((size_t)b * HID_ + h) * LK_ + k] = acc[r];
        }
    }
}

// ---------------------------------------------------------------------------
// Fused energy + mask + softmax.  (The 134M-tanh bottleneck kernel.)
// One block per (b,q); thread t handles key index k=t.
//   e[k] = sum_h tanh(Q[b,q,h] + KT[b,h,k]) * Wv[h]   (KT coalesced over k)
// Inner loop: 1 coalesced global load + shared b128 broadcasts + v_tanh_f32.
// ---------------------------------------------------------------------------
__global__ void energy_softmax(const float* __restrict__ Q, const float* __restrict__ KT,
                               const float* __restrict__ Wv, const uint8_t* __restrict__ mask,
                               float* __restrict__ attn) {
    const int bq = blockIdx.x;        // 0 .. B*LQ-1
    const int b  = bq >> 8;           // LQ_ == 256
    const int t  = threadIdx.x;       // key index, 0..255

    __shared__ __align__(16) float qs[HID_];
    __shared__ __align__(16) float wv[HID_];
    __shared__ float red[LK_];

    qs[t] = Q[(size_t)bq * HID_ + t];
    wv[t] = Wv[t];
    __syncthreads();

    const float* ktp = KT + (size_t)b * HID_ * LK_ + t;
    float e0 = 0.f, e1 = 0.f;
#pragma unroll 4
    for (int h = 0; h < HID_; h += 4) {
        const v4f q4 = *(const v4f*)(qs + h);   // ds_load_b128 broadcast
        const v4f w4 = *(const v4f*)(wv + h);   // ds_load_b128 broadcast
        const float k0 = ktp[(size_t)(h + 0) * LK_];
        const float k1 = ktp[(size_t)(h + 1) * LK_];
        const float k2 = ktp[(size_t)(h + 2) * LK_];
        const float k3 = ktp[(size_t)(h + 3) * LK_];
        e0 = fmaf(fast_tanhf(q4.x + k0), w4.x, e0);
        e1 = fmaf(fast_tanhf(q4.y + k1), w4.y, e1);
        e0 = fmaf(fast_tanhf(q4.z + k2), w4.z, e0);
        e1 = fmaf(fast_tanhf(q4.w + k3), w4.w, e1);
    }
    float e = e0 + e1;
    if (!mask[(size_t)bq * LK_ + t]) e = -1e10f;

    // block max
    red[t] = e; __syncthreads();
#pragma unroll
    for (int s = 128; s > 0; s >>= 1) {
        if (t < s) red[t] = fmaxf(red[t], red[t + s]);
        __syncthreads();
    }
    const float mx = red[0];
    __syncthreads();

    const float ex = __expf(e - mx);
    red[t] = ex; __syncthreads();
#pragma unroll
    for (int s = 128; s > 0; s >>= 1) {
        if (t < s) red[t] += red[t + s];
        __syncthreads();
    }
    attn[(size_t)bq * LK_ + t] = ex * (1.f / red[0]);
}

// ---------------------------------------------------------------------------
// expected[b,q,:] = attn[b,q,:] @ annotations[b,:,:]   (2048 x 256 x 512)
// One 16x16 tile per wave via V_WMMA_F32_16X16X4_F32.
// ---------------------------------------------------------------------------
__global__ void expected_gemm(const float* __restrict__ attn, const float* __restrict__ ann,
                              float* __restrict__ out) {
    const int lane = threadIdx.x & 31;
    const int wave = threadIdx.x >> 5;
    const int lo   = lane & 15;
    const int hi   = lane >> 4;
    const int row0 = blockIdx.x * 16;                 // global (b,q) row
    const int col0 = (blockIdx.y * 4 + wave) * 16;    // column into KSIZE
    const int b    = row0 >> 8;                       // LQ_ == 256

    const float* Arow = attn + (size_t)(row0 + lo) * LK_ + 2 * hi;
    const float* Bp   = ann + (size_t)b * LK_ * KSIZE_ + (size_t)(2 * hi) * KSIZE_ + col0 + lo;

    v8f acc = {};
#pragma unroll 8
    for (int k0 = 0; k0 < LK_; k0 += 4) {
        v2f a = *(const v2f*)(Arow + k0);
        v2f bb;
        bb.x = Bp[(size_t)k0 * KSIZE_];
        bb.y = Bp[(size_t)(k0 + 1) * KSIZE_];
        acc = __builtin_amdgcn_wmma_f32_16x16x4_f32(
            false, a, false, bb, (short)0, acc, false, false);
    }

    float* o = out + (size_t)(row0 + 8 * hi) * KSIZE_ + col0 + lo;
#pragma unroll
    for (int r = 0; r < 8; ++r) o[(size_t)r * KSIZE_] = acc[r];
}

// ---------------------------------------------------------------------------
extern "C" void kernel_launch(void* const* d_in, const int* in_sizes, int n_in,
                              void* d_out, int out_size, void* d_ws, size_t ws_size,
                              hipStream_t stream) {
    (void)in_sizes; (void)n_in; (void)out_size; (void)ws_size;

    const float*   hidden = (const float*)d_in[0];   // [B,LQ,QSIZE]
    const float*   ann    = (const float*)d_in[1];   // [B,LK,KSIZE]
    const uint8_t* mask   = (const uint8_t*)d_in[2]; // [B,LQ,LK] bool
    const float*   Wq     = (const float*)d_in[3];   // [QSIZE,HID]
    const float*   Wk     = (const float*)d_in[4];   // [KSIZE,HID]
    const float*   Wv     = (const float*)d_in[5];   // [HID]

    float* attn_out = (float*)d_out;                              // B*LQ*LK
    float* exp_out  = attn_out + (size_t)B_ * LQ_ * LK_;          // B*LQ*KSIZE

    float* Qws  = (float*)d_ws;                                   // B*LQ*HID (2 MB)
    float* KTws = Qws + (size_t)B_ * LQ_ * HID_;                  // B*HID*LK (2 MB)

    const dim3 blk(128);  // 4 waves -> 4 adjacent 16x16 tiles

    // Q = hidden @ W_q        (2048 x 512 x 256), normal layout
    proj_gemm<false><<<dim3((B_ * LQ_) / 16, HID_ / 64), blk, 0, stream>>>(
        hidden, Wq, Qws, QSIZE_, HID_);

    // KT = (annotations @ W_k)^T per batch  (stored [b][h][k])
    proj_gemm<true><<<dim3((B_ * LK_) / 16, HID_ / 64), blk, 0, stream>>>(
        ann, Wk, KTws, KSIZE_, HID_);

    // energy + mask + softmax -> attention weights (first output)
    energy_softmax<<<dim3(B_ * LQ_), dim3(LK_), 0, stream>>>(
        Qws, KTws, Wv, mask, attn_out);

    // expected_annotation = attn @ annotations (second output)
    expected_gemm<<<dim3((B_ * LQ_) / 16, KSIZE_ / 64), blk, 0, stream>>>(
        attn_out, ann, exp_out);
}